// CustomMultiLabelMarginLoss_54107997995579
// MI455X (gfx1250) — compile-verified
//
#include <hip/hip_runtime.h>

typedef __attribute__((ext_vector_type(2))) float v2f;
typedef __attribute__((ext_vector_type(8))) float v8f;

#define B_SIZE  64
#define C_SIZE  1536
#define NTILES  (C_SIZE / 16)   // 96 column tiles
#define MAXPOS  128             // data guarantees <= 128 positives per row

// One workgroup (8 wave32) per sample.
// Margin tile M[p][n] = 1 - x_p + x_n computed as WMMA rank-1 update:
//   D = A x B + C,  A[:,0] = -x_p (K=0 only), B[0,:] = 1, C[p][n] = 1 + x_n
// Then relu + negative-class mask + accumulate on the 8 C/D VGPRs.
__global__ __launch_bounds__(256)
void CustomMultiLabelMarginLoss_sample_kernel(const float* __restrict__ pred,
                                              const int*   __restrict__ target,
                                              float*       __restrict__ partial) {
    __shared__ float s_pred[C_SIZE];     // this sample's logits
    __shared__ float s_negmul[C_SIZE];   // 1.0 if class is negative, 0.0 if positive
    __shared__ float s_xp[MAXPOS];       // gathered positive logits (padded with 1e30)
    __shared__ int   s_first;            // index of first -1 in target row
    __shared__ float s_wavesum[8];

    const int row  = blockIdx.x;
    const int tid  = threadIdx.x;
    const int lane = tid & 31;
    const int wave = tid >> 5;

    const float* prow = pred   + row * C_SIZE;
    const int*   trow = target + row * C_SIZE;

    if (tid == 0) s_first = C_SIZE;
    __syncthreads();

    // Stage logits, init neg-mask, find first -1 (end of positive list).
    for (int c = tid; c < C_SIZE; c += 256) {
        s_pred[c]   = prow[c];
        s_negmul[c] = 1.0f;
        if (trow[c] == -1) atomicMin(&s_first, c);
    }
    __syncthreads();

    const int k = (s_first < MAXPOS) ? s_first : MAXPOS;  // number of positives

    // Mark positive classes (entries before the first -1).
    for (int c = tid; c < s_first; c += 256) {
        int t = trow[c];
        if (t >= 0 && t < C_SIZE) s_negmul[t] = 0.0f;
    }
    // Gather positive logits deterministically in target-list order; pad with
    // huge value so padded rows produce relu(1 - 1e30 + x_n) = 0.
    if (tid < MAXPOS) {
        float v = 1e30f;
        if (tid < k) {
            int t = trow[tid];
            if (t >= 0 && t < C_SIZE) v = s_pred[t];
        }
        s_xp[tid] = v;
    }
    __syncthreads();

    const int nptiles = (k + 15) >> 4;   // uniform across block -> EXEC stays full

    float lacc = 0.0f;

    // B matrix: K=0 row of ones (VGPR0, lanes 0-15); everything else zero.
    v2f bmat;
    bmat.x = (lane < 16) ? 1.0f : 0.0f;
    bmat.y = 0.0f;

    // Waves split the 96 column tiles.
    for (int nt = wave; nt < NTILES; nt += 8) {
        const int   col  = nt * 16 + (lane & 15);     // C/D column for this lane
        const float nm   = s_negmul[col];             // mask out positive n
        const float cval = 1.0f + s_pred[col];        // C[p][n] = 1 + x_n (all 8 rows)
        v8f cc;
#pragma unroll
        for (int r = 0; r < 8; ++r) cc[r] = cval;

        for (int pt = 0; pt < nptiles; ++pt) {
            // A matrix: K=0 column = -x_p (VGPR0, lanes 0-15 hold M=0..15).
            v2f amat;
            amat.x = (lane < 16) ? -s_xp[pt * 16 + lane] : 0.0f;
            amat.y = 0.0f;

            // D[p][n] = (-x_p)*1 + (1 + x_n)  -- emits v_wmma_f32_16x16x4_f32
            v8f dd = __builtin_amdgcn_wmma_f32_16x16x4_f32(
                false, amat, false, bmat, (short)0, cc, false, false);

#pragma unroll
            for (int r = 0; r < 8; ++r)
                lacc += fmaxf(dd[r], 0.0f) * nm;
        }
    }

    // Reduce across the wave (wave32), then across the 8 waves via LDS.
    for (int off = 16; off > 0; off >>= 1)
        lacc += __shfl_xor(lacc, off, 32);
    if (lane == 0) s_wavesum[wave] = lacc;
    __syncthreads();

    if (tid == 0) {
        float tot = 0.0f;
#pragma unroll
        for (int w = 0; w < 8; ++w) tot += s_wavesum[w];
        const float pos = (float)k;
        const float neg = (float)(C_SIZE - k);
        partial[row] = tot / (pos * neg);   // k in [8,128] -> denom > 0 always
    }
}

// Deterministic final mean over the 64 per-sample losses.
__global__ void CustomMultiLabelMarginLoss_final_kernel(const float* __restrict__ partial,
                                                        float* __restrict__ out) {
    if (threadIdx.x == 0 && blockIdx.x == 0) {
        float s = 0.0f;
        for (int i = 0; i < B_SIZE; ++i) s += partial[i];
        out[0] = s / (float)B_SIZE;
    }
}

extern "C" void kernel_launch(void* const* d_in, const int* in_sizes, int n_in,
                              void* d_out, int out_size, void* d_ws, size_t ws_size,
                              hipStream_t stream) {
    const float* pred   = (const float*)d_in[0];
    const int*   target = (const int*)  d_in[1];
    float*       out    = (float*)d_out;
    float*       part   = (float*)d_ws;   // 64 floats of scratch

    CustomMultiLabelMarginLoss_sample_kernel<<<B_SIZE, 256, 0, stream>>>(pred, target, part);
    CustomMultiLabelMarginLoss_final_kernel<<<1, 32, 0, stream>>>(part, out);
}